// HighOrderAggregator_17918603558961
// MI455X (gfx1250) — compile-verified
//
#include <hip/hip_runtime.h>

// ---------------------------------------------------------------------------
// HighOrderAggregator for MI455X (gfx1250, wave32, WMMA).
//   h0 = vecs; h1 = A*h0; h2 = A*h1        (SpMM via global_atomic_add_f32)
//   out = sum_o layernorm(relu(h_o @ W[o] + b[o])) * scale[o] + offset[o]
// GEMM uses V_WMMA_F32_16X16X4_F32 (full f32 precision; GEMM is not the
// bottleneck -- SpMM traffic is, and it lives mostly in the 192MB L2).
// ---------------------------------------------------------------------------

typedef float v2f __attribute__((ext_vector_type(2)));
typedef float v8f __attribute__((ext_vector_type(8)));

#define D 256            // DIN == DOUT == 256
#define LDS_STRIDE 260   // 256 + 4 pad -> conflict-free WMMA fragment reads

// ---------------------------------------------------------------- zero fill
__global__ void hoa_zero_kernel(float* __restrict__ p, long n) {
  long i = (long)blockIdx.x * blockDim.x + threadIdx.x;
  long stride = (long)gridDim.x * blockDim.x;
  for (; i < n; i += stride) p[i] = 0.0f;
}

// ------------------------------------------------------------------- SpMM
// One wave (32 lanes) per edge: gather x[col]*val, scatter-add into y[row].
// 256 floats/row = 2 float4 per lane; 8 f32 atomics per lane.
__global__ __launch_bounds__(256)
void hoa_spmm_kernel(const int* __restrict__ erow, const int* __restrict__ ecol,
                     const float* __restrict__ eval, const float* __restrict__ x,
                     float* __restrict__ y, int E) {
  int wid  = (blockIdx.x * blockDim.x + threadIdx.x) >> 5;
  int lane = threadIdx.x & 31;
  if (wid >= E) return;

  int   r = erow[wid];
  int   c = ecol[wid];
  float v = eval[wid];

  const float4* xr = (const float4*)(x + (size_t)c * D);
  float*        yr = y + (size_t)r * D;

  float4 x0 = xr[lane];
  float4 x1 = xr[lane + 32];
  int b0 = lane * 4;
  int b1 = (lane + 32) * 4;

  unsafeAtomicAdd(yr + b0 + 0, v * x0.x);
  unsafeAtomicAdd(yr + b0 + 1, v * x0.y);
  unsafeAtomicAdd(yr + b0 + 2, v * x0.z);
  unsafeAtomicAdd(yr + b0 + 3, v * x0.w);
  unsafeAtomicAdd(yr + b1 + 0, v * x1.x);
  unsafeAtomicAdd(yr + b1 + 1, v * x1.y);
  unsafeAtomicAdd(yr + b1 + 2, v * x1.z);
  unsafeAtomicAdd(yr + b1 + 3, v * x1.w);
}

// --------------------------------------------- fused GEMM + bias/relu/norm
// Block = 256 threads = 8 waves. Each block: 16 output rows x 256 cols.
// Wave w owns two 16x16 N-tiles at n0 = 32*w. K swept in 64 steps of 4
// with V_WMMA_F32_16X16X4_F32 (A from LDS, B from global -- W is 256KB, hot
// in cache). Epilogue: bias+relu -> LDS, wave32 shfl reduction for
// mean/var over the 256 columns, normalize, accumulate into out.
__global__ __launch_bounds__(256)
void hoa_gemm_norm_kernel(const float* __restrict__ X, const float* __restrict__ Wp,
                          const float* __restrict__ bias, const float* __restrict__ scl,
                          const float* __restrict__ offs, float* __restrict__ out,
                          int Nn, int accum) {
  __shared__ float sA[16 * LDS_STRIDE];   // reused for post-activation tile

  const int tid     = threadIdx.x;
  const int lane    = tid & 31;
  const int w       = tid >> 5;
  const int rowBase = blockIdx.x * 16;

  // ---- stage 16x256 A tile into LDS (coalesced, rows clamped for tails)
  for (int i = tid; i < 16 * D; i += 256) {
    int r = i >> 8;
    int c = i & 255;
    int gr = rowBase + r;
    if (gr >= Nn) gr = Nn - 1;
    sA[r * LDS_STRIDE + c] = X[(size_t)gr * D + c];
  }
  __syncthreads();

  // ---- WMMA fragment coordinates (ISA 16x16x4 f32 layout)
  const int m    = lane & 15;     // A row / B col within tile
  const int half = lane >> 4;     // 0: K=0,1   1: K=2,3
  const int n0   = w * 32;

  v8f c0 = {0.f,0.f,0.f,0.f,0.f,0.f,0.f,0.f};
  v8f c1 = {0.f,0.f,0.f,0.f,0.f,0.f,0.f,0.f};

#pragma unroll 4
  for (int kk = 0; kk < 64; ++kk) {
    const int ka = kk * 4 + 2 * half;

    v2f a;
    a.x = sA[m * LDS_STRIDE + ka];
    a.y = sA[m * LDS_STRIDE + ka + 1];

    const float* w0 = Wp + (size_t)ka * D + n0 + m;   // W[k][n], row-major
    const float* w1 = w0 + D;
    v2f bA; bA.x = w0[0];  bA.y = w1[0];
    v2f bB; bB.x = w0[16]; bB.y = w1[16];

    c0 = __builtin_amdgcn_wmma_f32_16x16x4_f32(false, a, false, bA,
                                               (short)0, c0, false, false);
    c1 = __builtin_amdgcn_wmma_f32_16x16x4_f32(false, a, false, bB,
                                               (short)0, c1, false, false);
  }
  __syncthreads();               // done reading sA as A-tile

  // ---- bias + relu, park activation tile in LDS
  const int col0 = n0 + m;
  const int col1 = col0 + 16;
  const float bv0 = bias[col0];
  const float bv1 = bias[col1];
#pragma unroll
  for (int r = 0; r < 8; ++r) {
    int rowT = r + 8 * half;     // D-matrix row mapping
    sA[rowT * LDS_STRIDE + col0] = fmaxf(c0[r] + bv0, 0.0f);
    sA[rowT * LDS_STRIDE + col1] = fmaxf(c1[r] + bv1, 0.0f);
  }
  __syncthreads();

  // ---- per-row layernorm: wave w handles rows 2w and 2w+1
  for (int rr = 0; rr < 2; ++rr) {
    int rowT = 2 * w + rr;
    float vals[8];
    float s = 0.0f, sq = 0.0f;
#pragma unroll
    for (int j = 0; j < 8; ++j) {
      float v = sA[rowT * LDS_STRIDE + lane + 32 * j];
      vals[j] = v;
      s  += v;
      sq += v * v;
    }
#pragma unroll
    for (int o = 16; o > 0; o >>= 1) {       // wave32 butterfly reduction
      s  += __shfl_xor(s,  o, 32);
      sq += __shfl_xor(sq, o, 32);
    }
    float mean = s  * (1.0f / 256.0f);
    float var  = sq * (1.0f / 256.0f) - mean * mean;  // biased (tf.nn.moments)
    float inv  = rsqrtf(var + 1e-9f);

    int gRow = rowBase + rowT;
    if (gRow < Nn) {
      float* op = out + (size_t)gRow * D;
#pragma unroll
      for (int j = 0; j < 8; ++j) {
        int c = lane + 32 * j;
        float o2 = scl[c] * (vals[j] - mean) * inv + offs[c];
        op[c] = accum ? (op[c] + o2) : o2;
      }
    }
  }
}

// ---------------------------------------------------------------------------
extern "C" void kernel_launch(void* const* d_in, const int* in_sizes, int n_in,
                              void* d_out, int out_size, void* d_ws, size_t ws_size,
                              hipStream_t stream) {
  const float* vecs   = (const float*)d_in[0];
  const int*   erow   = (const int*)  d_in[1];
  const int*   ecol   = (const int*)  d_in[2];
  const float* eval   = (const float*)d_in[3];
  const float* W      = (const float*)d_in[4];  // (3, 256, 256)
  const float* bias   = (const float*)d_in[5];  // (3, 256)
  const float* offset = (const float*)d_in[6];  // (3, 256)
  const float* scale  = (const float*)d_in[7];  // (3, 256)

  const int N = in_sizes[0] / D;   // 100000
  const int E = in_sizes[1];       // 3200000

  float* hop1 = (float*)d_ws;
  float* hop2 = hop1 + (size_t)N * D;
  float* out  = (float*)d_out;

  const long nd = (long)N * D;
  hoa_zero_kernel<<<4096, 256, 0, stream>>>(hop1, nd);
  hoa_zero_kernel<<<4096, 256, 0, stream>>>(hop2, nd);

  const int spmmBlocks = (E + 7) / 8;              // 8 waves (edges) / block
  hoa_spmm_kernel<<<spmmBlocks, 256, 0, stream>>>(erow, ecol, eval, vecs, hop1, E);
  hoa_spmm_kernel<<<spmmBlocks, 256, 0, stream>>>(erow, ecol, eval, hop1, hop2, E);

  const int rowBlocks = (N + 15) / 16;
  hoa_gemm_norm_kernel<<<rowBlocks, 256, 0, stream>>>(
      vecs, W,             bias,       scale,       offset,       out, N, 0);
  hoa_gemm_norm_kernel<<<rowBlocks, 256, 0, stream>>>(
      hop1, W + D * D,     bias + D,   scale + D,   offset + D,   out, N, 1);
  hoa_gemm_norm_kernel<<<rowBlocks, 256, 0, stream>>>(
      hop2, W + 2 * D * D, bias + 2*D, scale + 2*D, offset + 2*D, out, N, 1);
}